// RetroactiveNeuromodulation_84335977824715
// MI455X (gfx1250) — compile-verified
//
#include <hip/hip_runtime.h>
#include <hip/hip_bf16.h>

#define B_   256
#define OBS_ 128
#define ACT_ 16
#define HID_ 512
#define HCHUNKS_ 4
#define HCH_ (HID_ / HCHUNKS_)   // 128 h-rows per chunk

typedef __attribute__((ext_vector_type(2))) float v2f;
typedef __attribute__((ext_vector_type(4))) float v4f;
typedef __attribute__((ext_vector_type(8))) float v8f;

// ---------------------------------------------------------------------------
// Kernel 1: partial rec over an h-chunk:
//   rec_part[c][b][k] = sum_{h in chunk c} h_pre[b,h]*(W_hh[h,k]+alpha[h,k]*hebb[b,h,k])
// grid (B, 4) -> 1024 blocks of 128 threads; float4 coalesced streaming of hebb.
// ---------------------------------------------------------------------------
__global__ void rec_kernel(const float* __restrict__ h_pre,
                           const float* __restrict__ hebb,
                           const float* __restrict__ W_hh,
                           const float* __restrict__ alpha,
                           float* __restrict__ rec_part) {
    __shared__ float sh[HCH_];
    const int b     = blockIdx.x;
    const int chunk = blockIdx.y;             // 0..3
    const int h0    = chunk * HCH_;
    const int tid   = threadIdx.x;            // 0..127
    sh[tid] = h_pre[b * HID_ + h0 + tid];
    __syncthreads();

    const int k = tid * 4;                    // each thread owns 4 consecutive k
    const float* hb = hebb + (size_t)b * HID_ * HID_ + (size_t)h0 * HID_;
    const float* wp = W_hh  + (size_t)h0 * HID_;
    const float* ap = alpha + (size_t)h0 * HID_;
    v4f acc = {0.f, 0.f, 0.f, 0.f};
#pragma unroll 4
    for (int h = 0; h < HCH_; ++h) {
        const float hp = sh[h];
        const v4f w  = *(const v4f*)(wp + h * HID_ + k);
        const v4f al = *(const v4f*)(ap + h * HID_ + k);
        const v4f hv = *(const v4f*)(hb + h * HID_ + k);
        acc += hp * (w + al * hv);
    }
    *(v4f*)(rec_part + ((size_t)chunk * B_ + b) * HID_ + k) = acc;
}

// ---------------------------------------------------------------------------
// Kernel 2: h_post = tanh(obs @ i2h_w.T + i2h_b + sum_c rec_part[c])
// via V_WMMA_F32_16X16X4_F32.  M=256, N=512, K=128.
// One wave per 16x16 output tile; 4 waves per block.
// A 16x4 f32 fragment: lanes 0-15 -> M, VGPR0 = K {0 | 2}, VGPR1 = K {1 | 3}.
// B 4x16 f32 fragment: lanes hold N, VGPR0 = K {0 | 2}, VGPR1 = K {1 | 3}.
// C/D 16x16: VGPR r -> M = r (+8 for lanes 16-31), N = lane&15.
// ---------------------------------------------------------------------------
__global__ void i2h_wmma_kernel(const float* __restrict__ obs,
                                const float* __restrict__ i2h_w,
                                const float* __restrict__ i2h_b,
                                const float* __restrict__ rec_part,
                                float* __restrict__ h_post) {
    const int lane  = threadIdx.x & 31;
    const int wave  = threadIdx.x >> 5;       // 0..3
    const int tileN = blockIdx.x * 4 + wave;  // 0..31
    const int tileM = blockIdx.y;             // 0..15
    const int m0 = tileM * 16, n0 = tileN * 16;
    const int half = lane >> 4;               // 0: K+0/1, 1: K+2/3
    const int l    = lane & 15;

    const float* arow = obs   + (size_t)(m0 + l) * OBS_;   // A row  (M = l)
    const float* brow = i2h_w + (size_t)(n0 + l) * OBS_;   // B[k][n]=i2h_w[n][k]

    v8f acc = {};
    for (int k0 = 0; k0 < OBS_; k0 += 4) {
        const int kb = k0 + half * 2;
        v2f a, bf;
        a.x  = arow[kb];  a.y  = arow[kb + 1];
        bf.x = brow[kb];  bf.y = brow[kb + 1];
        acc = __builtin_amdgcn_wmma_f32_16x16x4_f32(
                  false, a, false, bf, (short)0, acc, false, false);
    }

    const int N = n0 + l;
    const float bias = i2h_b[N];
#pragma unroll
    for (int r = 0; r < 8; ++r) {
        const int M   = m0 + half * 8 + r;
        const size_t idx = (size_t)M * HID_ + N;
        const float rec = rec_part[idx]
                        + rec_part[(size_t)1 * B_ * HID_ + idx]
                        + rec_part[(size_t)2 * B_ * HID_ + idx]
                        + rec_part[(size_t)3 * B_ * HID_ + idx];
        h_post[idx] = tanhf(acc[r] + bias + rec);
    }
}

// ---------------------------------------------------------------------------
// Kernel 3: actor softmax, critic value, modulator m.  One wave32 per batch;
// 18 simultaneous dot products reduced with wave32 shuffles.
// ---------------------------------------------------------------------------
__global__ void head_kernel(const float* __restrict__ h_post,
                            const float* __restrict__ actor_w,
                            const float* __restrict__ actor_b,
                            const float* __restrict__ critic_w,
                            const float* __restrict__ critic_b,
                            const float* __restrict__ mod_w,
                            const float* __restrict__ mod_b,
                            float* __restrict__ act_prob,
                            float* __restrict__ value,
                            float* __restrict__ m_out) {
    const int b    = blockIdx.x;
    const int lane = threadIdx.x;             // 0..31 (wave32)
    float a[ACT_];
#pragma unroll
    for (int i = 0; i < ACT_; ++i) a[i] = 0.f;
    float c = 0.f, md = 0.f;

    for (int j = lane; j < HID_; j += 32) {
        const float h = h_post[b * HID_ + j];
#pragma unroll
        for (int i = 0; i < ACT_; ++i) a[i] += h * actor_w[i * HID_ + j];
        c  += h * critic_w[j];
        md += h * mod_w[j];
    }
#pragma unroll
    for (int off = 16; off > 0; off >>= 1) {
#pragma unroll
        for (int i = 0; i < ACT_; ++i) a[i] += __shfl_xor(a[i], off, 32);
        c  += __shfl_xor(c, off, 32);
        md += __shfl_xor(md, off, 32);
    }

    float logits[ACT_];
    float mx = -1e30f;
#pragma unroll
    for (int i = 0; i < ACT_; ++i) {
        logits[i] = a[i] + actor_b[i];
        mx = fmaxf(mx, logits[i]);
    }
    float s = 0.f;
#pragma unroll
    for (int i = 0; i < ACT_; ++i) { logits[i] = expf(logits[i] - mx); s += logits[i]; }
    const float inv = 1.f / s;
    if (lane < ACT_) act_prob[b * ACT_ + lane] = logits[lane] * inv;
    if (lane == 0) {
        value[b] = c + critic_b[0];
        m_out[b] = tanhf(md + mod_b[0]);
    }
}

// ---------------------------------------------------------------------------
// Kernel 4 (bandwidth king):
//   hebb_new  = clip(hebb + (m[b]*mf_w[k]+mf_b[k]) * trace, -2, 2)
//   trace_new = (1-eta)*trace + eta * h_pre[b,h] * h_post[b,k]
// Grid-stride over float4; NT loads/stores so the 4x268MB streams bypass
// L2 retention (keeps the small hot tables and hebb's pass-1 tail cached).
// ---------------------------------------------------------------------------
__global__ void plastic_kernel(const float* __restrict__ hebb,
                               const float* __restrict__ trace,
                               const float* __restrict__ h_pre,
                               const float* __restrict__ h_post,
                               const float* __restrict__ m_in,
                               const float* __restrict__ mf_w,
                               const float* __restrict__ mf_b,
                               const float* __restrict__ eta_p,
                               float* __restrict__ hebb_new,
                               float* __restrict__ trace_new) {
    const float eta  = eta_p[0];
    const float omet = 1.f - eta;
    const size_t total4 = (size_t)B_ * HID_ * HID_ / 4;
    const size_t stride = (size_t)gridDim.x * blockDim.x;
    for (size_t i4 = (size_t)blockIdx.x * blockDim.x + threadIdx.x;
         i4 < total4; i4 += stride) {
        const int    k4  = (int)(i4 & (HID_ / 4 - 1));   // i4 % 128
        const size_t row = i4 >> 7;                      // i4 / 128
        const int    h   = (int)(row & (HID_ - 1));
        const int    b   = (int)(row >> 9);
        const int    k   = k4 * 4;

        const float mb = m_in[b];
        const float hp = h_pre[b * HID_ + h];
        const v4f mfw = *(const v4f*)(mf_w + k);
        const v4f mfb = *(const v4f*)(mf_b + k);
        const v4f hpo = *(const v4f*)(h_post + b * HID_ + k);

        const v4f hb = __builtin_nontemporal_load((const v4f*)hebb  + i4);
        const v4f tr = __builtin_nontemporal_load((const v4f*)trace + i4);

        const v4f modf = mb * mfw + mfb;
        v4f hn = hb + modf * tr;
        hn.x = fminf(fmaxf(hn.x, -2.f), 2.f);
        hn.y = fminf(fmaxf(hn.y, -2.f), 2.f);
        hn.z = fminf(fmaxf(hn.z, -2.f), 2.f);
        hn.w = fminf(fmaxf(hn.w, -2.f), 2.f);
        const v4f tn = omet * tr + (eta * hp) * hpo;

        __builtin_nontemporal_store(hn, (v4f*)hebb_new  + i4);
        __builtin_nontemporal_store(tn, (v4f*)trace_new + i4);
    }
}

// ---------------------------------------------------------------------------
extern "C" void kernel_launch(void* const* d_in, const int* in_sizes, int n_in,
                              void* d_out, int out_size, void* d_ws, size_t ws_size,
                              hipStream_t stream) {
    const float* obs      = (const float*)d_in[0];
    const float* h_pre    = (const float*)d_in[1];
    const float* hebb     = (const float*)d_in[2];
    const float* trace    = (const float*)d_in[3];
    const float* i2h_w    = (const float*)d_in[4];
    const float* i2h_b    = (const float*)d_in[5];
    const float* W_hh     = (const float*)d_in[6];
    const float* alpha    = (const float*)d_in[7];
    const float* eta      = (const float*)d_in[8];
    const float* mf_w     = (const float*)d_in[9];
    const float* mf_b     = (const float*)d_in[10];
    const float* actor_w  = (const float*)d_in[11];
    const float* actor_b  = (const float*)d_in[12];
    const float* critic_w = (const float*)d_in[13];
    const float* critic_b = (const float*)d_in[14];
    const float* mod_w    = (const float*)d_in[15];
    const float* mod_b    = (const float*)d_in[16];

    float* out       = (float*)d_out;
    float* act_prob  = out;                                   // [B, ACT]
    float* value     = act_prob + (size_t)B_ * ACT_;          // [B, 1]
    float* m_out     = value + B_;                            // [B, 1]
    float* h_post    = m_out + B_;                            // [B, HID]
    float* hebb_new  = h_post + (size_t)B_ * HID_;            // [B, HID, HID]
    float* trace_new = hebb_new + (size_t)B_ * HID_ * HID_;   // [B, HID, HID]

    float* rec_part = (float*)d_ws;       // [HCHUNKS, B, HID] = 2 MB scratch

    rec_kernel<<<dim3(B_, HCHUNKS_), 128, 0, stream>>>(h_pre, hebb, W_hh, alpha,
                                                       rec_part);
    i2h_wmma_kernel<<<dim3(8, 16), 128, 0, stream>>>(obs, i2h_w, i2h_b,
                                                     rec_part, h_post);
    head_kernel<<<B_, 32, 0, stream>>>(h_post, actor_w, actor_b, critic_w,
                                       critic_b, mod_w, mod_b,
                                       act_prob, value, m_out);
    plastic_kernel<<<8192, 256, 0, stream>>>(hebb, trace, h_pre, h_post, m_out,
                                             mf_w, mf_b, eta,
                                             hebb_new, trace_new);
}